// AdExNeuron_90314572300838
// MI455X (gfx1250) — compile-verified
//
#include <hip/hip_runtime.h>

// Problem shape from the reference: B=32, T=2000, D=1024.
#define B_   32
#define T_   2000
#define D_   1024
#define N_   (B_ * D_)     // 32768 neurons, one thread each
#define BLK  256
#define GRP  4             // timesteps per pipeline group
#define NGRP (T_ / GRP)    // 500

// ---- CDNA5 async global->LDS path (guarded; falls back to register prefetch)
#if defined(__HIP_DEVICE_COMPILE__) && defined(__gfx1250__) && \
    __has_builtin(__builtin_amdgcn_global_load_async_to_lds_b32) && \
    __has_builtin(__builtin_amdgcn_s_wait_asynccnt)
#define USE_ASYNC 1
#else
#define USE_ASYNC 0
#endif

#if USE_ASYNC
typedef __attribute__((address_space(1))) int as1_int;   // global (prints as "__device__ int *")
typedef __attribute__((address_space(3))) int as3_int;   // LDS    (prints as "__shared__ int *")
#endif

struct AdExP {
    float E_L, V_T, inv_dT, Delta_T, R, a, bb, V_reset, V_spike, k, c2;
};

// One AdEx step. Matches reference update order exactly:
//   dV uses old V; w update uses NEW V; spike check on new V; reset / w+b last.
__device__ __forceinline__ float adex_step(float& V, float& w, float it, const AdExP& p) {
    float ex = p.Delta_T * __expf((V - p.V_T) * p.inv_dT);        // v_exp_f32 (TRANS)
    V = V + p.k * ((p.E_L - V) + ex + p.R * (it - w));            // V += dt/tau_m * dV
    w = w + p.c2 * (p.a * (V - p.E_L) - w);                       // w += dt/tau_w * dw
    float spk = (V >= p.V_spike) ? 1.0f : 0.0f;                   // v_cmp + v_cndmask
    V = (spk > 0.0f) ? p.V_reset : V;
    w = (spk > 0.0f) ? (w + p.bb) : w;
    return spk;
}

__global__ __launch_bounds__(BLK)
void AdEx_stream_kernel(const float* __restrict__ I,
                        const float* __restrict__ prm,
                        float* __restrict__ out) {
    const int tid = threadIdx.x;
    const int n   = blockIdx.x * BLK + tid;       // neuron id in [0, N_)
    const int b   = n >> 10;                      // n / D_
    const int d   = n & (D_ - 1);                 // n % D_

    // Uniform scalar loads of the 11 params.
    AdExP p;
    const float tau_m = prm[0];
    p.E_L = prm[1];  p.V_T = prm[2];  p.Delta_T = prm[3];
    p.R   = prm[4];
    const float tau_w = prm[5];
    p.a   = prm[6];  p.bb  = prm[7];  p.V_reset = prm[8];
    p.V_spike = prm[9];
    const float dt = prm[10];
    p.k      = dt / tau_m;
    p.c2     = dt / tau_w;
    p.inv_dT = 1.0f / p.Delta_T;

    float V = p.E_L;
    float w = 0.0f;

    const int base = (b * T_) * D_ + d;           // fits in int (max ~63.5M)
    const float* __restrict__ gin  = I   + base;  // + t*D_ per step
    float*       __restrict__ gout = out + base;

#if USE_ASYNC
    // Double-buffered async pipeline: 2 buffers x 4 timesteps x 4B per thread.
    __shared__ __align__(16) float sbuf[2][BLK][GRP];   // 8 KB of the 320 KB LDS

    // Prologue: 8 async B32 loads in flight (groups 0 and 1).
    #pragma unroll
    for (int j = 0; j < GRP; ++j)
        __builtin_amdgcn_global_load_async_to_lds_b32(
            (as1_int*)(float*)(gin + j * D_),
            (as3_int*)(&sbuf[0][tid][j]), 0, 0);
    #pragma unroll
    for (int j = 0; j < GRP; ++j)
        __builtin_amdgcn_global_load_async_to_lds_b32(
            (as1_int*)(float*)(gin + (GRP + j) * D_),
            (as3_int*)(&sbuf[1][tid][j]), 0, 0);

    for (int g = 0; g < NGRP; ++g) {
        const int cur = g & 1;
        // 8 outstanding -> wait until the oldest 4 (current buffer) have landed.
        __builtin_amdgcn_s_wait_asynccnt(4);
        asm volatile("" ::: "memory");            // compiler ordering fence only

        const float4 iv = *(const float4*)(&sbuf[cur][tid][0]);   // ds_load_b128

        const int t0 = g * GRP;
        float s0 = adex_step(V, w, iv.x, p);
        float s1 = adex_step(V, w, iv.y, p);
        float s2 = adex_step(V, w, iv.z, p);
        float s3 = adex_step(V, w, iv.w, p);

        __builtin_nontemporal_store(s0, gout + (t0 + 0) * D_);
        __builtin_nontemporal_store(s1, gout + (t0 + 1) * D_);
        __builtin_nontemporal_store(s2, gout + (t0 + 2) * D_);
        __builtin_nontemporal_store(s3, gout + (t0 + 3) * D_);

        asm volatile("" ::: "memory");
        // Refill this buffer two groups ahead; clamp address near the end so the
        // outstanding-count stays uniform (extra loads read already-consumed data).
        int tn = t0 + 2 * GRP;
        if (tn > T_ - GRP) tn = T_ - GRP;
        #pragma unroll
        for (int j = 0; j < GRP; ++j)
            __builtin_amdgcn_global_load_async_to_lds_b32(
                (as1_int*)(float*)(gin + (tn + j) * D_),
                (as3_int*)(&sbuf[cur][tid][j]), 0, 0);
    }
#else
    // Fallback: register double-buffer, one-group-ahead NT prefetch.
    float c0 = __builtin_nontemporal_load(gin + 0 * D_);
    float c1 = __builtin_nontemporal_load(gin + 1 * D_);
    float c2 = __builtin_nontemporal_load(gin + 2 * D_);
    float c3 = __builtin_nontemporal_load(gin + 3 * D_);

    for (int g = 0; g < NGRP; ++g) {
        int tn = g * GRP + GRP;
        if (tn > T_ - GRP) tn = T_ - GRP;
        const float n0 = __builtin_nontemporal_load(gin + (tn + 0) * D_);
        const float n1 = __builtin_nontemporal_load(gin + (tn + 1) * D_);
        const float n2 = __builtin_nontemporal_load(gin + (tn + 2) * D_);
        const float n3 = __builtin_nontemporal_load(gin + (tn + 3) * D_);

        const int t0 = g * GRP;
        float s0 = adex_step(V, w, c0, p);
        float s1 = adex_step(V, w, c1, p);
        float s2 = adex_step(V, w, c2, p);
        float s3 = adex_step(V, w, c3, p);

        __builtin_nontemporal_store(s0, gout + (t0 + 0) * D_);
        __builtin_nontemporal_store(s1, gout + (t0 + 1) * D_);
        __builtin_nontemporal_store(s2, gout + (t0 + 2) * D_);
        __builtin_nontemporal_store(s3, gout + (t0 + 3) * D_);

        c0 = n0; c1 = n1; c2 = n2; c3 = n3;
    }
#endif
}

extern "C" void kernel_launch(void* const* d_in, const int* in_sizes, int n_in,
                              void* d_out, int out_size, void* d_ws, size_t ws_size,
                              hipStream_t stream) {
    (void)in_sizes; (void)n_in; (void)out_size; (void)d_ws; (void)ws_size;
    const float* I      = (const float*)d_in[0];   // (B, T, D) fp32
    const float* params = (const float*)d_in[1];   // 11 fp32 scalars
    float*       out    = (float*)d_out;           // (B, T, D) fp32 spikes

    dim3 grid(N_ / BLK);   // 128 workgroups x 256 threads = 32768 threads (8 waves/WG)
    dim3 block(BLK);
    AdEx_stream_kernel<<<grid, block, 0, stream>>>(I, params, out);
}